// Net_35519379537969
// MI455X (gfx1250) — compile-verified
//
#include <hip/hip_runtime.h>
#include <math.h>

typedef float v2f __attribute__((ext_vector_type(2)));
typedef float v8f __attribute__((ext_vector_type(8)));

// ---------------- degree / norm ----------------

__global__ void deg_kernel(const long long* __restrict__ dst, float* __restrict__ deg, int E) {
    int e = blockIdx.x * blockDim.x + threadIdx.x;
    if (e < E) atomicAdd(&deg[(int)dst[e]], 1.0f);
}

// in-place deg -> dinv ; +1.0f accounts for the self-loop, so deg+1 >= 1 always
__global__ void dinv_kernel(float* __restrict__ deg, int n) {
    int i = blockIdx.x * blockDim.x + threadIdx.x;
    if (i < n) deg[i] = rsqrtf(deg[i] + 1.0f);
}

// ---------------- dense GEMM 1: h = x @ W1  ([N,128] x [128,32]) via V_WMMA_F32_16X16X4_F32 ----

__global__ void gemm1_wmma(const float* __restrict__ x, const float* __restrict__ W,
                           float* __restrict__ h, int nTiles, int N) {
    int wave = (blockIdx.x * blockDim.x + threadIdx.x) >> 5;
    int lane = threadIdx.x & 31;
    if (wave >= nTiles) return;                 // wave-uniform exit: EXEC stays all-1s for WMMA
    int row0 = wave * 16;
    int m  = lane & 15;                         // A: row M ; B/C/D: column N
    int kg = lane >> 4;                         // half-wave K group
    int rr = row0 + m; if (rr > N - 1) rr = N - 1;
    const float* xr = x + (size_t)rr * 128;

    v8f acc0 = {}; v8f acc1 = {};
    for (int k = 0; k < 128; k += 4) {
        int kk = k + 2 * kg;
        v2f a;  a.x  = xr[kk];                a.y  = xr[kk + 1];
        v2f b0; b0.x = W[kk * 32 + m];        b0.y = W[(kk + 1) * 32 + m];
        v2f b1; b1.x = W[kk * 32 + 16 + m];   b1.y = W[(kk + 1) * 32 + 16 + m];
        acc0 = __builtin_amdgcn_wmma_f32_16x16x4_f32(false, a, false, b0, (short)0, acc0, false, false);
        acc1 = __builtin_amdgcn_wmma_f32_16x16x4_f32(false, a, false, b1, (short)0, acc1, false, false);
    }
    #pragma unroll
    for (int v = 0; v < 8; ++v) {
        int r = row0 + v + 8 * kg;              // C/D layout: M = v + 8*(lane/16)
        if (r < N) {
            h[(size_t)r * 32 + m]      = acc0[v];
            h[(size_t)r * 32 + 16 + m] = acc1[v];
        }
    }
}

// ---------------- edge scatter, layer 1 (32 features, float4 per thread) ----------------

__global__ void scatter1_kernel(const long long* __restrict__ src, const long long* __restrict__ dst,
                                const float* __restrict__ dinv, const float* __restrict__ h,
                                float* __restrict__ agg, int E) {
    int t = blockIdx.x * blockDim.x + threadIdx.x;
    int e = t >> 3;
    if (e >= E) return;
    int q = (t & 7) * 4;
    int s = (int)src[e], d = (int)dst[e];
    float nrm = dinv[s] * dinv[d];
    float4 v = *(const float4*)(h + (size_t)s * 32 + q);
    float* ap = agg + (size_t)d * 32 + q;
    atomicAdd(ap + 0, v.x * nrm);
    atomicAdd(ap + 1, v.y * nrm);
    atomicAdd(ap + 2, v.z * nrm);
    atomicAdd(ap + 3, v.w * nrm);
}

// ---------------- self-loop + bias + ReLU (in place into h) ----------------

__global__ void bias_relu_kernel(float* __restrict__ h, const float* __restrict__ agg,
                                 const float* __restrict__ dinv, const float* __restrict__ b1, int n) {
    int t = blockIdx.x * blockDim.x + threadIdx.x;
    if (t >= n) return;
    int node = t >> 5;
    int f = t & 31;
    float di = dinv[node];
    float v = agg[t] + h[t] * di * di + b1[f];
    h[t] = v > 0.0f ? v : 0.0f;
}

// ---------------- dense GEMM 2: oh = h1 @ W2 ([N,32] x [32,8]) ----------------

__global__ void gemm2_wmma(const float* __restrict__ h1, const float* __restrict__ W2,
                           float* __restrict__ oh, int nTiles, int N) {
    int wave = (blockIdx.x * blockDim.x + threadIdx.x) >> 5;
    int lane = threadIdx.x & 31;
    if (wave >= nTiles) return;
    int row0 = wave * 16;
    int m  = lane & 15;
    int kg = lane >> 4;
    int rr = row0 + m; if (rr > N - 1) rr = N - 1;
    const float* xr = h1 + (size_t)rr * 32;

    v8f acc = {};
    for (int k = 0; k < 32; k += 4) {
        int kk = k + 2 * kg;
        v2f a; a.x = xr[kk]; a.y = xr[kk + 1];
        float bx = 0.0f, by = 0.0f;
        if (m < 8) { bx = W2[kk * 8 + m]; by = W2[(kk + 1) * 8 + m]; }
        v2f b; b.x = bx; b.y = by;
        acc = __builtin_amdgcn_wmma_f32_16x16x4_f32(false, a, false, b, (short)0, acc, false, false);
    }
    if (m < 8) {
        #pragma unroll
        for (int v = 0; v < 8; ++v) {
            int r = row0 + v + 8 * kg;
            if (r < N) oh[(size_t)r * 8 + m] = acc[v];
        }
    }
}

// ---------------- edge scatter, layer 2 (8 features, float4 per thread) ----------------

__global__ void scatter2_kernel(const long long* __restrict__ src, const long long* __restrict__ dst,
                                const float* __restrict__ dinv, const float* __restrict__ oh,
                                float* __restrict__ agg2, int E) {
    int t = blockIdx.x * blockDim.x + threadIdx.x;
    int e = t >> 1;
    if (e >= E) return;
    int q = (t & 1) * 4;
    int s = (int)src[e], d = (int)dst[e];
    float nrm = dinv[s] * dinv[d];
    float4 v = *(const float4*)(oh + (size_t)s * 8 + q);
    float* ap = agg2 + (size_t)d * 8 + q;
    atomicAdd(ap + 0, v.x * nrm);
    atomicAdd(ap + 1, v.y * nrm);
    atomicAdd(ap + 2, v.z * nrm);
    atomicAdd(ap + 3, v.w * nrm);
}

// ---------------- self-loop + bias + log_softmax ----------------

__global__ void finalize_kernel(const float* __restrict__ agg2, const float* __restrict__ oh,
                                const float* __restrict__ dinv, const float* __restrict__ b2,
                                float* __restrict__ out, int N) {
    int i = blockIdx.x * blockDim.x + threadIdx.x;
    if (i >= N) return;
    float di = dinv[i];
    float d2 = di * di;
    float v[8];
    float mx = -INFINITY;
    #pragma unroll
    for (int f = 0; f < 8; ++f) {
        v[f] = agg2[(size_t)i * 8 + f] + oh[(size_t)i * 8 + f] * d2 + b2[f];
        mx = fmaxf(mx, v[f]);
    }
    float s = 0.0f;
    #pragma unroll
    for (int f = 0; f < 8; ++f) s += expf(v[f] - mx);
    float l = logf(s);
    #pragma unroll
    for (int f = 0; f < 8; ++f) out[(size_t)i * 8 + f] = v[f] - mx - l;
}

// ---------------- launcher ----------------

static inline size_t align256(size_t s) { return (s + 255) & ~(size_t)255; }

extern "C" void kernel_launch(void* const* d_in, const int* in_sizes, int n_in,
                              void* d_out, int out_size, void* d_ws, size_t ws_size,
                              hipStream_t stream) {
    const float*     x   = (const float*)d_in[0];
    const long long* ei  = (const long long*)d_in[1];   // int64 [2,E]
    const float*     W1  = (const float*)d_in[2];
    const float*     b1  = (const float*)d_in[3];
    const float*     W2  = (const float*)d_in[4];
    const float*     b2  = (const float*)d_in[5];

    const int N = in_sizes[0] / 128;
    const int E = in_sizes[1] / 2;
    const long long* src = ei;
    const long long* dst = ei + E;

    // workspace layout (agg1 region recycled for oh/agg2 after ReLU)
    char* ws = (char*)d_ws;
    size_t off = 0;
    float* dinv = (float*)(ws + off); off += align256((size_t)N * 4);
    float* h    = (float*)(ws + off); off += align256((size_t)N * 32 * 4);
    float* agg1 = (float*)(ws + off); off += align256((size_t)N * 32 * 4);
    float* oh   = agg1;              // N*8 floats
    float* agg2 = agg1 + (size_t)N * 8;

    (void)ws_size; (void)n_in; (void)out_size;

    // degree + norm
    hipMemsetAsync(dinv, 0, (size_t)N * 4, stream);
    hipMemsetAsync(agg1, 0, (size_t)N * 32 * 4, stream);
    deg_kernel<<<(E + 255) / 256, 256, 0, stream>>>(dst, dinv, E);
    dinv_kernel<<<(N + 255) / 256, 256, 0, stream>>>(dinv, N);

    // layer 1
    int tiles = (N + 15) / 16;
    int gemmBlocks = (tiles * 32 + 127) / 128;          // 4 waves per 128-thread block
    gemm1_wmma<<<gemmBlocks, 128, 0, stream>>>(x, W1, h, tiles, N);
    scatter1_kernel<<<(E * 8 + 255) / 256, 256, 0, stream>>>(src, dst, dinv, h, agg1, E);
    bias_relu_kernel<<<(N * 32 + 255) / 256, 256, 0, stream>>>(h, agg1, dinv, b1, N * 32);

    // layer 2 (oh/agg2 reuse agg1's storage; agg1 values are dead after bias_relu)
    gemm2_wmma<<<gemmBlocks, 128, 0, stream>>>(h, W2, oh, tiles, N);
    hipMemsetAsync(agg2, 0, (size_t)N * 8 * 4, stream);
    scatter2_kernel<<<(E * 2 + 255) / 256, 256, 0, stream>>>(src, dst, dinv, oh, agg2, E);

    // bias + log_softmax
    finalize_kernel<<<(N + 255) / 256, 256, 0, stream>>>(agg2, oh, dinv, b2, (float*)d_out, N);
}